// WKPooling_38663295599198
// MI455X (gfx1250) — compile-verified
//
#include <hip/hip_runtime.h>

// CDNA5 (gfx1250) WK-Pooling.
// Per-token 9x9 Gram G = T*T^T (T = 9x768 f32) via V_WMMA_F32_16X16X4_F32,
// fed from LDS that is filled with GLOBAL_LOAD_ASYNC_TO_LDS_B128 (ASYNCcnt).
// One b128 LDS load per lane feeds two WMMAs (dual accumulators to break the
// D->C chain). Tiny per-token math (Cholesky replacement for QR, alpha,
// cosine-variance) on a few lanes; variance-weighted reduction in kernel 2.

typedef float v2f __attribute__((ext_vector_type(2)));
typedef float v4f __attribute__((ext_vector_type(4)));
typedef float v8f __attribute__((ext_vector_type(8)));

#define NUM_LAYERS 13
#define LAYER_START 4
#define NL 9            // layers used
#define BATCH 32
#define SEQ 128
#define NTOK (SEQ - 1)  // max unmask
#define DIM 768
#define STOKP 772       // padded LDS row stride (772 % 64 == 4 -> spreads banks)
#define EPS 1e-8f

__global__ __launch_bounds__(128)
void wk_token_kernel(const float* __restrict__ hs, const int* __restrict__ mask,
                     float* __restrict__ embBuf, float* __restrict__ varBuf)
{
    __shared__ float stok[16 * STOKP];   // 16 rows (9 real + 7 zero pad) x 768
    __shared__ float gpart[4 * 256];     // per-wave 16x16 partial Grams
    __shared__ float G[81];              // 9x9 Gram
    __shared__ float aK[NL], nK[NL], alpha[NL];
    __shared__ int   s_um;

    const int tid = threadIdx.x;
    const int bid = blockIdx.x;
    const int b = bid / NTOK;
    const int n = bid % NTOK;

    if (tid == 0) {
        int s = 0;
        for (int i = 0; i < SEQ; ++i) s += mask[i];   // reference uses row 0 only
        s_um = (s - 1 < NTOK) ? (s - 1) : NTOK;
    }
    __syncthreads();
    const int um = s_um;
    if (n >= um) {
        if (tid == 0) varBuf[b * NTOK + n] = 0.0f;    // zero weight -> no contribution
        return;
    }

    // ---- Stage A: async global -> LDS stream of the 9x768 slab (b128/lane) ----
    {
        const long long strideL = (long long)BATCH * SEQ * DIM;          // floats
        const long long base =
            ((long long)LAYER_START * BATCH + b) * (long long)(SEQ * DIM) +
            (long long)n * DIM;
        const uint64_t sbase = (uint64_t)(const void*)(hs + base);       // SGPR pair
        const uint32_t ldsStokBase = (uint32_t)(uintptr_t)(&stok[0]);    // LDS byte off
        const uint32_t strideLBytes = (uint32_t)(strideL * 4);

        for (int idx = tid; idx < NL * (DIM / 4); idx += 128) {          // 1728 xfers
            const int l  = idx / (DIM / 4);
            const int d4 = idx - l * (DIM / 4);
            const uint32_t voff = (uint32_t)l * strideLBytes + (uint32_t)d4 * 16u;
            const uint32_t dsa  = ldsStokBase + ((uint32_t)(l * STOKP + d4 * 4)) * 4u;
            asm volatile("global_load_async_to_lds_b128 %0, %1, %2"
                         :: "v"(dsa), "v"(voff), "s"(sbase) : "memory");
        }
        // zero pad rows 9..15 (plain LDS stores)
        for (int idx = tid; idx < (16 - NL) * DIM; idx += 128) {
            int l = NL + idx / DIM, d = idx % DIM;
            stok[l * STOKP + d] = 0.0f;
        }
        asm volatile("s_wait_asynccnt 0x0" ::: "memory");
    }
    __syncthreads();

    // ---- Stage B: Gram via f32 WMMA; contiguous 192-wide K span per wave ----
    {
        const int wave = tid >> 5;
        const int lane = tid & 31;
        const int row  = lane & 15;   // M (== N for symmetric Gram)
        const int kh   = lane >> 4;   // upper-half lanes carry the K+4.. columns
        const int kbase = wave * (DIM / 4);   // 192 floats of K per wave
        v8f c0 = {}, c1 = {};
        #pragma unroll 4
        for (int t = 0; t < 24; ++t) {
            // One b128 load feeds two WMMAs:
            //   wmma0 K-cols = kbase + {8t, 8t+1, 8t+4, 8t+5}
            //   wmma1 K-cols = kbase + {8t+2, 8t+3, 8t+6, 8t+7}
            v4f q = *reinterpret_cast<const v4f*>(
                        &stok[row * STOKP + kbase + t * 8 + kh * 4]);
            v2f a0 = {q.x, q.y};
            v2f a1 = {q.z, q.w};
            // Symmetric Gram: A(16x4) and B(4x16)=A^T share per-lane contents.
            c0 = __builtin_amdgcn_wmma_f32_16x16x4_f32(false, a0, false, a0,
                                                       (short)0, c0, false, false);
            c1 = __builtin_amdgcn_wmma_f32_16x16x4_f32(false, a1, false, a1,
                                                       (short)0, c1, false, false);
        }
        // C/D layout: lane L, VGPR v holds D[M = v + 8*(L>>4)][N = L&15]
        #pragma unroll
        for (int v = 0; v < 8; ++v)
            gpart[wave * 256 + (v + 8 * kh) * 16 + row] = c0[v] + c1[v];
    }
    __syncthreads();
    if (tid < 81) {
        int i = tid / 9, j = tid - i * 9;
        G[tid] = gpart[0 * 256 + i * 16 + j] + gpart[1 * 256 + i * 16 + j]
               + gpart[2 * 256 + i * 16 + j] + gpart[3 * 256 + i * 16 + j];
    }
    __syncthreads();

    // ---- Stage C: per-k scalars via Cholesky of the window Gram ----
    if (tid < NL) {
        const int k = tid;
        int idx[5]; int m = 0;
        if (k >= 2)     { idx[m++] = k - 2; idx[m++] = k - 1; }  // left (empty if k<2)
        if (k + 1 < NL)   idx[m++] = k + 1;                      // right
        if (k + 2 < NL)   idx[m++] = k + 2;
        idx[m++] = k;                                            // self last

        // Upper Cholesky: R^T R = G[idx][idx]  (row-sign-equivalent to QR's R)
        float R[5][5];
        for (int j = 0; j < m; ++j) {
            for (int i = 0; i < j; ++i) {
                float s = G[idx[i] * 9 + idx[j]];
                for (int p = 0; p < i; ++p) s -= R[p][i] * R[p][j];
                R[i][j] = s / R[i][i];
            }
            float s = G[idx[j] * 9 + idx[j]];
            for (int p = 0; p < j; ++p) s -= R[p][j] * R[p][j];
            R[j][j] = sqrtf(s);
        }
        // alignment score
        float cn[4];
        for (int j = 0; j < m - 1; ++j) {
            float s2 = 0.0f;
            for (int p = 0; p <= j; ++p) s2 += R[p][j] * R[p][j];
            cn[j] = sqrtf(s2);
        }
        float num = 0.0f, hr2 = 0.0f;
        for (int i = 0; i < m - 1; ++i) {
            float mv = 0.0f;
            for (int j = i; j < m - 1; ++j) mv += R[i][j] / cn[j];  // R[i][j]=0 for j<i
            mv /= (float)(m - 1);
            num += mv * R[i][m - 1];
            hr2 += R[i][m - 1] * R[i][m - 1];
        }
        float a = num / sqrtf(hr2);
        a = 1.0f / (a * (float)m * 2.0f);
        float r2 = hr2 + R[m - 1][m - 1] * R[m - 1][m - 1];
        float nv = fabsf(R[m - 1][m - 1]) / sqrtf(r2);
        aK[k] = a; nK[k] = nv;
    }
    if (tid == NL) {
        // variance of adjacent-layer cosine similarity (ddof=1 over 8 values)
        float dsim[NL - 1]; float sum = 0.0f;
        for (int i = 0; i < NL - 1; ++i) {
            float wprod = sqrtf(G[i * 9 + i]) * sqrtf(G[(i + 1) * 9 + (i + 1)]);
            float sim = G[(i + 1) * 9 + i] / fmaxf(wprod, EPS);
            dsim[i] = sim; sum += sim;
        }
        float mean = sum / (float)(NL - 1);
        float var = 0.0f;
        for (int i = 0; i < NL - 1; ++i) { float t = dsim[i] - mean; var += t * t; }
        varBuf[b * NTOK + n] = var / (float)(NL - 2);
    }
    __syncthreads();
    if (tid == 0) {
        float sa = 0.0f, sn = 0.0f;
        for (int k = 0; k < NL; ++k) { sa += aK[k]; sn += nK[k]; }
        float t = 0.0f;
        for (int k = 0; k < NL; ++k) { float al = aK[k] / sa + nK[k] / sn; alpha[k] = al; t += al; }
        for (int k = 0; k < NL; ++k) alpha[k] /= t;
    }
    __syncthreads();

    // ---- Stage D: token_emb = alpha @ tok ----
    float* dst = &embBuf[((long long)b * NTOK + n) * DIM];
    for (int d = tid; d < DIM; d += 128) {
        float acc = 0.0f;
        #pragma unroll
        for (int l = 0; l < NL; ++l) acc += alpha[l] * stok[l * STOKP + d];
        dst[d] = acc;
    }
}

__global__ __launch_bounds__(256)
void wk_reduce_kernel(const int* __restrict__ mask, const float* __restrict__ embBuf,
                      const float* __restrict__ varBuf, float* __restrict__ out)
{
    __shared__ float svar[NTOK];
    __shared__ float sinv;
    __shared__ int   s_um;
    const int tid = threadIdx.x;
    const int b = blockIdx.x;

    if (tid == 0) {
        int s = 0;
        for (int i = 0; i < SEQ; ++i) s += mask[i];
        s_um = (s - 1 < NTOK) ? (s - 1) : NTOK;
    }
    __syncthreads();
    const int um = s_um;
    if (tid < NTOK) svar[tid] = (tid < um) ? varBuf[b * NTOK + tid] : 0.0f;
    __syncthreads();
    if (tid == 0) {
        float s = 0.0f;
        for (int i = 0; i < um; ++i) s += svar[i];
        sinv = 1.0f / s;
    }
    __syncthreads();
    const float inv = sinv;
    const float* eb = &embBuf[(long long)b * NTOK * DIM];
    if (tid < DIM / 4) {                      // 192 active lanes, b128 traffic
        const int d4 = tid;
        v4f acc = {};
        for (int nn = 0; nn < um; ++nn) {
            v4f e = *reinterpret_cast<const v4f*>(&eb[(long long)nn * DIM + d4 * 4]);
            const float w = svar[nn];
            acc.x += e.x * w; acc.y += e.y * w;
            acc.z += e.z * w; acc.w += e.w * w;
        }
        acc.x *= inv; acc.y *= inv; acc.z *= inv; acc.w *= inv;
        *reinterpret_cast<v4f*>(&out[b * DIM + d4 * 4]) = acc;
    }
}

extern "C" void kernel_launch(void* const* d_in, const int* in_sizes, int n_in,
                              void* d_out, int out_size, void* d_ws, size_t ws_size,
                              hipStream_t stream)
{
    (void)in_sizes; (void)n_in; (void)out_size; (void)ws_size;
    const float* hs   = (const float*)d_in[0];   // (13,32,128,768) f32
    const int*   mask = (const int*)d_in[1];     // (32,128) i32
    float* out    = (float*)d_out;               // (32,768) f32
    float* embBuf = (float*)d_ws;                                     // 32*127*768 f32
    float* varBuf = embBuf + (size_t)BATCH * NTOK * DIM;              // 32*127 f32

    wk_token_kernel<<<BATCH * NTOK, 128, 0, stream>>>(hs, mask, embBuf, varBuf);
    wk_reduce_kernel<<<BATCH, 256, 0, stream>>>(mask, embBuf, varBuf, out);
}